// Postprocess_24575802867982
// MI455X (gfx1250) — compile-verified
//
#include <hip/hip_runtime.h>
#include <hip/hip_bf16.h>

// ---------------- problem constants ----------------
constexpr int   N_PRED   = 102000;
constexpr int   K_TOP    = 2048;
constexpr int   NBINS    = 8192;   // 13-bit radix bins on sortable score key
constexpr int   CAP      = 8192;   // candidate list capacity == bitonic sort size
constexpr float MIN_CONF = 0.25f;
constexpr float MIN_IOU  = 0.45f;
constexpr float EPS_IOU  = 1e-7f;

// ---------------- workspace layout (bytes) ----------------
constexpr size_t OFF_BOXES = 0;                                   // N*4 f32 (ltrb, scaled+rounded)
constexpr size_t OFF_CONF  = OFF_BOXES + (size_t)N_PRED * 16;     // N f32
constexpr size_t OFF_CLS   = OFF_CONF  + (size_t)N_PRED * 4;      // N i32
constexpr size_t OFF_VALID = OFF_CLS   + (size_t)N_PRED * 4;      // N u8 (N % 4 == 0)
constexpr size_t OFF_KEY   = OFF_VALID + (size_t)N_PRED;          // N u32 sortable key
constexpr size_t OFF_HIST  = OFF_KEY   + (size_t)N_PRED * 4;      // NBINS u32
constexpr size_t OFF_CNT   = OFF_HIST  + (size_t)NBINS * 4;       // 4 u32: [selCount, threshBin, -, -]
constexpr size_t OFF_LIST  = OFF_CNT   + 16;                      // CAP u64 (8B aligned)
constexpr size_t OFF_TBOX  = OFF_LIST  + (size_t)CAP * 8;         // K*4 f32 (16B aligned)
constexpr size_t OFF_TCONF = OFF_TBOX  + (size_t)K_TOP * 16;      // K f32
constexpr size_t OFF_TCLS  = OFF_TCONF + (size_t)K_TOP * 4;       // K i32
constexpr size_t OFF_TVAL  = OFF_TCLS  + (size_t)K_TOP * 4;       // K u32
constexpr size_t OFF_SUP   = OFF_TVAL  + (size_t)K_TOP * 4;       // K * 64 u32 suppression bit-matrix
constexpr size_t OFF_ALIVE = OFF_SUP   + (size_t)K_TOP * 64 * 4;  // 64 u32

typedef float v2f __attribute__((ext_vector_type(2)));
typedef float v8f __attribute__((ext_vector_type(8)));

// sortable-ascending mapping of float (handles the -1.0 sentinel correctly)
__device__ __forceinline__ unsigned sortable_f32(float s) {
  unsigned b = __float_as_uint(s);
  return b ^ ((b & 0x80000000u) ? 0xFFFFFFFFu : 0x80000000u);
}

// ---------------- generic zero fill ----------------
__global__ void zero_kernel(unsigned* p, int n) {
  int i = blockIdx.x * blockDim.x + threadIdx.x;
  if (i < n) p[i] = 0u;
}

// ---------------- phase 1: decode (wave-per-row) ----------------
__global__ void decode_kernel(const float* __restrict__ pred,
                              const int* __restrict__ fh, const int* __restrict__ fw,
                              float* __restrict__ boxesAll, float* __restrict__ confAll,
                              int* __restrict__ clsAll, unsigned char* __restrict__ validAll,
                              unsigned* __restrict__ keyAll) {
  const int lane = threadIdx.x & 31;
  const int row  = blockIdx.x * (blockDim.x >> 5) + (threadIdx.x >> 5);
  if (row >= N_PRED) return;
  const float* r = pred + (size_t)row * 85;

  // prefetch the tail of this row's cachelines (emits global_prefetch_b8)
  __builtin_prefetch(r + 64, 0, 3);

  // argmax over 80 class scores: lanes cover c, c+32, c+64(half). Ties -> lowest idx.
  float c0 = r[5 + lane];
  float c1 = r[37 + lane];
  float c2 = (lane < 16) ? r[69 + lane] : -1e30f;
  float bv = c0; int bi = lane;
  if (c1 > bv) { bv = c1; bi = lane + 32; }
  if (c2 > bv) { bv = c2; bi = lane + 64; }
  #pragma unroll
  for (int off = 16; off > 0; off >>= 1) {
    float ov = __shfl_xor(bv, off, 32);
    int   oi = __shfl_xor(bi, off, 32);
    if (ov > bv || (ov == bv && oi < bi)) { bv = ov; bi = oi; }
  }

  if (lane == 0) {
    float x = r[0], y = r[1], w = r[2], h = r[3], obj = r[4];
    float conf = obj * bv;
    bool vclass = (bi < 8) && ((0xAFu >> bi) & 1u);   // {0,1,2,3,5,7}
    bool valid  = (conf > MIN_CONF) && vclass;
    float sx = (float)(*fw) / 640.0f;
    float sy = (float)(*fh) / 640.0f;
    float4 b;
    b.x = rintf((x - w * 0.5f) * sx);   // round-half-even == jnp.round
    b.y = rintf((y - h * 0.5f) * sy);
    b.z = rintf((x + w * 0.5f) * sx);
    b.w = rintf((y + h * 0.5f) * sy);
    ((float4*)boxesAll)[row] = b;
    confAll[row]  = conf;
    clsAll[row]   = bi;
    validAll[row] = valid ? 1 : 0;
    keyAll[row]   = sortable_f32(valid ? conf : -1.0f);
  }
}

// ---------------- phase 2: radix select + bitonic sort ----------------
__global__ void hist_kernel(const unsigned* __restrict__ keyAll, unsigned* __restrict__ hist) {
  int i = blockIdx.x * blockDim.x + threadIdx.x;
  if (i < N_PRED) atomicAdd(&hist[keyAll[i] >> 19], 1u);
}

__global__ void select_thresh_kernel(const unsigned* __restrict__ hist, unsigned* __restrict__ counters) {
  unsigned acc = 0, T = 0;
  for (int b = NBINS - 1; b >= 0; --b) {
    acc += hist[b];
    if (acc >= (unsigned)K_TOP) { T = (unsigned)b; break; }
  }
  counters[1] = T;
}

__global__ void collect_kernel(const unsigned* __restrict__ keyAll,
                               unsigned* __restrict__ counters,
                               unsigned long long* __restrict__ list) {
  int i = blockIdx.x * blockDim.x + threadIdx.x;
  if (i >= N_PRED) return;
  unsigned u = keyAll[i];
  if ((u >> 19) >= counters[1]) {
    unsigned pos = atomicAdd(&counters[0], 1u);
    if (pos < (unsigned)CAP)
      list[pos] = ((unsigned long long)(~u) << 32) | (unsigned)i;   // asc sort => score desc, idx asc
  }
}

__global__ void __launch_bounds__(1024)
sort_gather_kernel(const unsigned long long* __restrict__ list,
                   const unsigned* __restrict__ counters,
                   const float* __restrict__ boxesAll, const float* __restrict__ confAll,
                   const int* __restrict__ clsAll, const unsigned char* __restrict__ validAll,
                   float* __restrict__ topBoxes, float* __restrict__ topConf,
                   int* __restrict__ topCls, unsigned* __restrict__ topValid) {
  __shared__ unsigned long long s[CAP];   // 64 KB in 320 KB CDNA5 LDS
  const int tid = threadIdx.x;
  const unsigned M = min(counters[0], (unsigned)CAP);
  for (int t = tid; t < CAP; t += 1024) s[t] = (t < (int)M) ? list[t] : ~0ull;
  __syncthreads();
  for (int k = 2; k <= CAP; k <<= 1) {
    for (int j = k >> 1; j > 0; j >>= 1) {
      for (int t = tid; t < CAP; t += 1024) {
        int ixj = t ^ j;
        if (ixj > t) {
          unsigned long long a = s[t], b = s[ixj];
          bool up = ((t & k) == 0);
          if ((a > b) == up) { s[t] = b; s[ixj] = a; }
        }
      }
      __syncthreads();
    }
  }
  for (int t = tid; t < K_TOP; t += 1024) {
    unsigned idx = (unsigned)(s[t] & 0xFFFFFFFFu);
    if (idx < (unsigned)N_PRED) {
      ((float4*)topBoxes)[t] = ((const float4*)boxesAll)[idx];
      topConf[t]  = confAll[idx];
      topCls[t]   = clsAll[idx];
      topValid[t] = validAll[idx];
    } else {   // defensive: fewer than K candidates (cannot happen: all N rows keyed)
      ((float4*)topBoxes)[t] = make_float4(0.f, 0.f, 0.f, 0.f);
      topConf[t] = -1.0f; topCls[t] = 0; topValid[t] = 0u;
    }
  }
}

// ---------------- phase 3a: pairwise suppression bit-matrix via WMMA ----------------
// Each wave owns a 16x16 IoU tile. area_i + area_j outer-sum is a rank-2 matmul:
//   A(16x4) rows = [a_i, 1, 0, 0], B(4x16) cols = [1, a_j, 0, 0]^T  =>  D = a_i + a_j
// Box tile is DMA'd into LDS with GLOBAL_LOAD_ASYNC_TO_LDS_B128 (ASYNCcnt),
// no VGPR round-trip (CDNA5 ISA).
__global__ void __launch_bounds__(256)
sup_kernel(const float* __restrict__ topBoxes, unsigned* __restrict__ supM) {
  __shared__ float4 sb[K_TOP];   // 32 KB
  __shared__ float  sa[K_TOP];   //  8 KB

  // async DMA global -> LDS: 8 x b128 per thread, tracked by ASYNCcnt
  for (int t = threadIdx.x; t < K_TOP; t += blockDim.x) {
    unsigned ldsOff = (unsigned)(uintptr_t)(&sb[t]);               // low 32 bits == LDS offset
    unsigned long long gaddr = (unsigned long long)(uintptr_t)(topBoxes + (size_t)t * 4);
    asm volatile("global_load_async_to_lds_b128 %0, %1, off"
                 :: "v"(ldsOff), "v"(gaddr) : "memory");
  }
  asm volatile("s_wait_asynccnt 0x0" ::: "memory");
  __syncthreads();
  for (int t = threadIdx.x; t < K_TOP; t += blockDim.x) {
    float4 b = sb[t];
    sa[t] = (b.z - b.x) * (b.w - b.y);
  }
  __syncthreads();

  const int lane = threadIdx.x & 31;
  const int tile = blockIdx.x * (blockDim.x >> 5) + (threadIdx.x >> 5);  // 0..16383 exact
  const int rt = tile >> 7, ct = tile & 127;
  const int half = lane >> 4, l16 = lane & 15;

  // A 16x4 layout: lanes 0-15 hold K0(v0),K1(v1); lanes 16-31 hold K2,K3 (zero).
  v2f A, B;
  A.x = half ? 0.0f : sa[rt * 16 + l16];
  A.y = half ? 0.0f : 1.0f;
  // B 4x16 layout: v0 lanes0-15 = row K0 (ones), lanes16-31 = row K1 (a_j); v1 = rows K2,K3 = 0.
  B.x = half ? sa[ct * 16 + l16] : 1.0f;
  B.y = 0.0f;
  v8f C = {};
  v8f D = __builtin_amdgcn_wmma_f32_16x16x4_f32(false, A, false, B, (short)0, C, false, false);

  const float4 bj = sb[ct * 16 + l16];
  const int wordIdx = ct >> 1;
  const int shift   = (ct & 1) * 16;
  #pragma unroll
  for (int r = 0; r < 8; ++r) {
    int gi = rt * 16 + r + half * 8;                   // D row for this lane-half
    float4 bi = sb[gi];
    float ltx = fmaxf(bi.x, bj.x), lty = fmaxf(bi.y, bj.y);
    float rbx = fminf(bi.z, bj.z), rby = fminf(bi.w, bj.w);
    float inter = fmaxf(rbx - ltx, 0.0f) * fmaxf(rby - lty, 0.0f);
    float iou = inter / (D[r] - inter + EPS_IOU);      // D[r] = a_i + a_j
    int gj = ct * 16 + l16;
    bool sup = (iou > MIN_IOU) && (gj > gi);
    unsigned mask = __builtin_amdgcn_ballot_w32(sup);  // [15:0]=row gi(half0), [31:16]=row gi(half1)
    if (lane == 0)  atomicOr(&supM[(size_t)(rt * 16 + r) * 64 + wordIdx], (mask & 0xFFFFu) << shift);
    if (lane == 16) atomicOr(&supM[(size_t)(rt * 16 + r + 8) * 64 + wordIdx], (mask >> 16) << shift);
  }
}

// ---------------- phase 3b: sequential greedy scan (single wave, bitmask) ----------------
__global__ void scan_kernel(const unsigned* __restrict__ topValid,
                            const unsigned* __restrict__ supM,
                            unsigned* __restrict__ aliveOut) {
  const int lane = threadIdx.x;   // one wave32
  unsigned a0 = 0, a1 = 0;        // lane owns alive words [lane] and [lane+32]
  for (int b = 0; b < 32; ++b) {
    a0 |= (topValid[lane * 32 + b]        ? 1u : 0u) << b;
    a1 |= (topValid[(lane + 32) * 32 + b] ? 1u : 0u) << b;
  }
  for (int i = 0; i < K_TOP; ++i) {
    int w = i >> 5;
    unsigned word = (w < 32) ? a0 : a1;          // w uniform
    unsigned aw = __shfl((int)word, w & 31, 32);
    if ((aw >> (i & 31)) & 1u) {                 // alive[i] -> suppress its row (bits already j>i only)
      const unsigned* row = supM + (size_t)i * 64;
      a0 &= ~row[lane];
      a1 &= ~row[lane + 32];
    }
  }
  aliveOut[lane]      = a0;
  aliveOut[lane + 32] = a1;
}

// ---------------- phase 4: masked outputs ----------------
__global__ void finalize_kernel(const float* __restrict__ topBoxes, const float* __restrict__ topConf,
                                const int* __restrict__ topCls, const unsigned* __restrict__ alive,
                                float* __restrict__ out) {
  int k = blockIdx.x * blockDim.x + threadIdx.x;
  if (k >= K_TOP) return;
  unsigned keep = (alive[k >> 5] >> (k & 31)) & 1u;
  float m = keep ? 1.0f : 0.0f;
  float4 b = ((const float4*)topBoxes)[k];
  out[k * 4 + 0] = b.x * m;
  out[k * 4 + 1] = b.y * m;
  out[k * 4 + 2] = b.z * m;
  out[k * 4 + 3] = b.w * m;
  out[4 * K_TOP + k]  = topConf[k] * m;                       // out_conf
  out[5 * K_TOP + k]  = keep ? (float)topCls[k] : -1.0f;      // out_class_ids
  out[6 * K_TOP + k]  = m;                                    // keep
}

// ---------------- host ----------------
extern "C" void kernel_launch(void* const* d_in, const int* in_sizes, int n_in,
                              void* d_out, int out_size, void* d_ws, size_t ws_size,
                              hipStream_t stream) {
  const float* pred = (const float*)d_in[0];
  const int*   fh   = (const int*)d_in[1];
  const int*   fw   = (const int*)d_in[2];
  char* ws = (char*)d_ws;

  float*              boxesAll = (float*)(ws + OFF_BOXES);
  float*              confAll  = (float*)(ws + OFF_CONF);
  int*                clsAll   = (int*)(ws + OFF_CLS);
  unsigned char*      validAll = (unsigned char*)(ws + OFF_VALID);
  unsigned*           keyAll   = (unsigned*)(ws + OFF_KEY);
  unsigned*           hist     = (unsigned*)(ws + OFF_HIST);
  unsigned*           counters = (unsigned*)(ws + OFF_CNT);
  unsigned long long* list     = (unsigned long long*)(ws + OFF_LIST);
  float*              topBoxes = (float*)(ws + OFF_TBOX);
  float*              topConf  = (float*)(ws + OFF_TCONF);
  int*                topCls   = (int*)(ws + OFF_TCLS);
  unsigned*           topValid = (unsigned*)(ws + OFF_TVAL);
  unsigned*           supM     = (unsigned*)(ws + OFF_SUP);
  unsigned*           alive    = (unsigned*)(ws + OFF_ALIVE);

  // zero hist+counters (contiguous) and suppression matrix
  {
    int n = NBINS + 4;
    zero_kernel<<<(n + 255) / 256, 256, 0, stream>>>(hist, n);
    int m = K_TOP * 64;
    zero_kernel<<<(m + 255) / 256, 256, 0, stream>>>(supM, m);
  }
  // decode: 8 waves (rows) per 256-thread block
  decode_kernel<<<(N_PRED + 7) / 8, 256, 0, stream>>>(pred, fh, fw, boxesAll, confAll,
                                                      clsAll, validAll, keyAll);
  hist_kernel<<<(N_PRED + 255) / 256, 256, 0, stream>>>(keyAll, hist);
  select_thresh_kernel<<<1, 1, 0, stream>>>(hist, counters);
  collect_kernel<<<(N_PRED + 255) / 256, 256, 0, stream>>>(keyAll, counters, list);
  sort_gather_kernel<<<1, 1024, 0, stream>>>(list, counters, boxesAll, confAll, clsAll,
                                             validAll, topBoxes, topConf, topCls, topValid);
  // 128x128 tiles of 16x16, one wave each: 16384 waves = 2048 blocks of 8 waves
  sup_kernel<<<2048, 256, 0, stream>>>(topBoxes, supM);
  scan_kernel<<<1, 32, 0, stream>>>(topValid, supM, alive);
  finalize_kernel<<<(K_TOP + 255) / 256, 256, 0, stream>>>(topBoxes, topConf, topCls, alive,
                                                           (float*)d_out);
  (void)in_sizes; (void)n_in; (void)out_size; (void)ws_size;
}